// ForwardShift_18013092840173
// MI455X (gfx1250) — compile-verified
//
#include <hip/hip_runtime.h>
#include <stdint.h>

// Problem constants (reference: B=4, C=3, H=720, W=1280)
#define B_  4
#define C_  3
#define H_  720
#define W_  1280
#define HW_ (H_ * W_)          // 921600
#define R_  4

// Tile geometry: 32x8 source pixels per block; LDS accumulator covers
// footprint + flow halo of +-(HX-R)=6 pixels of flow displacement.
#define TW_  32
#define TH_  8
#define HX_  10
#define HY_  10
#define LW_  (TW_ + 2 * HX_)   // 52
#define LH_  (TH_ + 2 * HY_)   // 28
#define LCELLS_ (LW_ * LH_)    // 1456 cells x 4 planes = 23.3 KB LDS

typedef __attribute__((ext_vector_type(4))) unsigned int v4u;
typedef __attribute__((ext_vector_type(8))) int          v8i;
typedef __attribute__((ext_vector_type(4))) int          v4i;

// ---------------------------------------------------------------------------
// TDM: async 3D tile load (planes x rows x width, f32) -> LDS, contiguous
// [z][y][x] in LDS. Descriptor per CDNA5 ISA 8.3-8.6. clang-23 6-arg builtin.
// ---------------------------------------------------------------------------
__device__ __forceinline__ void tdm_load_3d(uint32_t lds_off, const void* gptr,
                                            uint32_t width, uint32_t rows,
                                            uint32_t planes,
                                            uint64_t row_stride_elems,
                                            uint64_t plane_stride_elems) {
  uint64_t ga = (uint64_t)(uintptr_t)gptr;
  v4u g0;
  g0[0] = 1u;                                                // count=1, user mode
  g0[1] = lds_off;                                           // lds_addr (bytes)
  g0[2] = (uint32_t)ga;                                      // global_addr[31:0]
  g0[3] = (uint32_t)((ga >> 32) & 0x01FFFFFFu) | (2u << 30); // addr[56:32] | type=2
  v8i g1;
  g1[0] = (int)(2u << 16);                                   // data_size = 4B
  g1[1] = (int)((width & 0xFFFFu) << 16);                    // tensor_dim0 lo16
  g1[2] = (int)(((width >> 16) & 0xFFFFu) |                  // tensor_dim0 hi16
                ((rows & 0xFFFFu) << 16));                   // tensor_dim1 lo16
  g1[3] = (int)(((rows >> 16) & 0xFFFFu) |                   // tensor_dim1 hi16
                ((width & 0xFFFFu) << 16));                  // tile_dim0
  g1[4] = (int)((rows & 0xFFFFu) |                           // tile_dim1
                ((planes & 0xFFFFu) << 16));                 // tile_dim2
  g1[5] = (int)(uint32_t)row_stride_elems;                   // dim0_stride[31:0]
  g1[6] = (int)((uint32_t)((row_stride_elems >> 32) & 0xFFFFu) | // dim0_stride[47:32]
                ((uint32_t)(plane_stride_elems & 0xFFFFu) << 16)); // dim1_stride[15:0]
  g1[7] = (int)(uint32_t)((plane_stride_elems >> 16) & 0xFFFFFFFFu); // dim1_stride[47:16]
  v4i g2 = {(int)planes, 0, 0, 0};   // tensor_dim2 (OOB bound for z); rest unused
  v4i g3 = {0, 0, 0, 0};
  v8i z8 = {0, 0, 0, 0, 0, 0, 0, 0};
  __builtin_amdgcn_tensor_load_to_lds(g0, g1, g2, g3, z8, 0);
}

// ---------------------------------------------------------------------------
// Pass 0: zero the whole output (weighted sums + weight sums), float4 stores.
// ---------------------------------------------------------------------------
__global__ void __launch_bounds__(256) fsplat_zero(float4* __restrict__ p) {
  p[(size_t)blockIdx.x * 256 + threadIdx.x] = make_float4(0.f, 0.f, 0.f, 0.f);
}

// ---------------------------------------------------------------------------
// Pass 1: scatter. Block = 32x8 source tile. TDM stages flow(2 planes) and
// src(3 planes) tiles into LDS. Pixels splat their 9x9 Gaussian footprint into
// an LDS accumulator tile with ds_add_f32; the tile is flushed once to global
// with HW f32 atomics. Outlier flows (|u|,|v| > 6) take a direct-global path.
// ---------------------------------------------------------------------------
__global__ void __launch_bounds__(256) fsplat_scatter(const float* __restrict__ src,
                                                      const float* __restrict__ flow,
                                                      float* __restrict__ out) {
  __shared__ float sacc[4 * LCELLS_];   // [w | c0 | c1 | c2] accumulator planes
  __shared__ float sstage[5 * 256];     // [u | v | c0 | c1 | c2] staged tiles

  const int tid = threadIdx.x;
  int t = blockIdx.x;
  const int x0 = (t % (W_ / TW_)) * TW_;  t /= (W_ / TW_);
  const int y0 = (t % (H_ / TH_)) * TH_;  t /= (H_ / TH_);
  const int b  = t;

  if (tid < 32) {                       // wave 0 issues both async tile loads
    uint32_t lds_base = (uint32_t)(uintptr_t)(&sstage[0]);
    tdm_load_3d(lds_base,
                flow + (size_t)b * 2 * HW_ + (size_t)y0 * W_ + x0,
                TW_, TH_, 2u, (uint64_t)W_, (uint64_t)HW_);
    tdm_load_3d(lds_base + 2u * 256u * sizeof(float),
                src + (size_t)b * C_ * HW_ + (size_t)y0 * W_ + x0,
                TW_, TH_, 3u, (uint64_t)W_, (uint64_t)HW_);
  }
  // Zero LDS accumulators while the TDM streams the staging region.
  for (int i = tid; i < 4 * LCELLS_; i += 256) sacc[i] = 0.0f;
  if (tid < 32) __builtin_amdgcn_s_wait_tensorcnt(0);
  __syncthreads();

  // Launder staging pointer: TDM wrote it behind the compiler's back.
  const float* sp = sstage;
  asm volatile("" : "+v"(sp));

  const int   lx = tid & (TW_ - 1);
  const int   ly = tid >> 5;
  const int   x  = x0 + lx;
  const int   y  = y0 + ly;
  const float u  = sp[tid];
  const float v  = sp[256 + tid];
  const float s0 = sp[512 + tid];
  const float s1 = sp[768 + tid];
  const float s2 = sp[1024 + tid];

  const float px = (float)x + u;
  const float py = (float)y + v;
  const int   bx = (int)floorf(px);
  const int   by = (int)floorf(py);

  // Separable x-direction weights: 9 exp instead of 81.
  float wx[9], ax2[9];
  int   txs[9];
  bool  vx[9];
#pragma unroll
  for (int i = 0; i < 9; ++i) {
    int   tx = bx + i - R_;
    float ad = (float)tx - px;
    float a2 = ad * ad;
    txs[i] = tx;
    ax2[i] = a2;
    wx[i]  = __expf(-a2);              // 2*sigma^2 == 1
    vx[i]  = (tx >= 0) & (tx < W_);
  }

  float* accw = out + (size_t)B_ * C_ * HW_ + (size_t)b * HW_;  // weight plane
  float* acc0 = out + ((size_t)b * C_ + 0) * HW_;
  float* acc1 = acc0 + HW_;
  float* acc2 = acc1 + HW_;

  const bool fast = (bx >= x0 - (HX_ - R_)) && (bx <= x0 + TW_ - 1 + (HX_ - R_)) &&
                    (by >= y0 - (HY_ - R_)) && (by <= y0 + TH_ - 1 + (HY_ - R_));

  if (fast) {
    const int lco = x0 - HX_;          // LDS tile column/row origin
    const int lro = y0 - HY_;
#pragma unroll 1
    for (int j = 0; j < 9; ++j) {
      int ty = by + j - R_;
      if (ty < 0 || ty >= H_) continue;
      float ad  = (float)ty - py;
      float ay2 = ad * ad;
      float wy  = __expf(-ay2);
      int   rowl = (ty - lro) * LW_ - lco;
#pragma unroll
      for (int i = 0; i < 9; ++i) {
        if (!vx[i]) continue;
        float d2 = ax2[i] + ay2;
        if (d2 > 16.0f) continue;      // radius^2 cull (matches d2 <= r2)
        float w    = wx[i] * wy;
        int   lidx = rowl + txs[i];
        atomicAdd(&sacc[lidx], w);                    // ds_add_f32
        atomicAdd(&sacc[LCELLS_ + lidx], w * s0);
        atomicAdd(&sacc[2 * LCELLS_ + lidx], w * s1);
        atomicAdd(&sacc[3 * LCELLS_ + lidx], w * s2);
      }
    }
  } else {                             // rare outlier flow: direct global path
#pragma unroll 1
    for (int j = 0; j < 9; ++j) {
      int ty = by + j - R_;
      if (ty < 0 || ty >= H_) continue;
      float ad  = (float)ty - py;
      float ay2 = ad * ad;
      float wy  = __expf(-ay2);
      int   row = ty * W_;
#pragma unroll
      for (int i = 0; i < 9; ++i) {
        if (!vx[i]) continue;
        float d2 = ax2[i] + ay2;
        if (d2 > 16.0f) continue;
        float w   = wx[i] * wy;
        int   idx = row + txs[i];
        unsafeAtomicAdd(accw + idx, w);               // global_atomic_add_f32
        unsafeAtomicAdd(acc0 + idx, w * s0);
        unsafeAtomicAdd(acc1 + idx, w * s1);
        unsafeAtomicAdd(acc2 + idx, w * s2);
      }
    }
  }
  __syncthreads();

  // Flush LDS accumulator tile to global: one HW atomic per touched cell/plane.
#pragma unroll 1
  for (int i = tid; i < LCELLS_; i += 256) {
    int r  = i / LW_;
    int c  = i - r * LW_;
    int gy = y0 - HY_ + r;
    int gx = x0 - HX_ + c;
    if (gy < 0 || gy >= H_ || gx < 0 || gx >= W_) continue;
    float w = sacc[i];
    if (w == 0.0f) continue;           // untouched halo cell
    int gidx = gy * W_ + gx;
    unsafeAtomicAdd(accw + gidx, w);
    unsafeAtomicAdd(acc0 + gidx, sacc[LCELLS_ + i]);
    unsafeAtomicAdd(acc1 + gidx, sacc[2 * LCELLS_ + i]);
    unsafeAtomicAdd(acc2 + gidx, sacc[3 * LCELLS_ + i]);
  }
}

// ---------------------------------------------------------------------------
// Pass 2: normalize in place. img /= (w + eps); mask = (w > 0).
// ---------------------------------------------------------------------------
__global__ void __launch_bounds__(256) fsplat_norm(float* __restrict__ out) {
  const int i = blockIdx.x * 256 + threadIdx.x;   // 0 .. B*HW-1
  float* accw = out + (size_t)B_ * C_ * HW_;
  const float w     = accw[i];
  const float denom = w + 1e-8f;
  const int   b  = i / HW_;
  const int   hw = i - b * HW_;
  const size_t base = ((size_t)b * C_) * HW_ + hw;
  out[base]           = out[base] / denom;
  out[base + HW_]     = out[base + HW_] / denom;
  out[base + 2 * HW_] = out[base + 2 * HW_] / denom;
  accw[i] = (w > 0.0f) ? 1.0f : 0.0f;
}

extern "C" void kernel_launch(void* const* d_in, const int* in_sizes, int n_in,
                              void* d_out, int out_size, void* d_ws, size_t ws_size,
                              hipStream_t stream) {
  const float* src  = (const float*)d_in[0];   // (B,C,H,W) f32
  const float* flow = (const float*)d_in[1];   // (B,2,H,W) f32
  float* out = (float*)d_out;                  // [img (B,C,H,W) | mask (B,1,H,W)]

  const int total   = B_ * C_ * HW_ + B_ * HW_;            // 14,745,600 floats
  const int zblocks = total / (256 * 4);                   // exact: 14,400
  const int sblocks = B_ * (H_ / TH_) * (W_ / TW_);        // 4*90*40 = 14,400
  const int nblocks = (B_ * HW_) / 256;                    // exact: 14,400

  fsplat_zero<<<zblocks, 256, 0, stream>>>((float4*)out);
  fsplat_scatter<<<sblocks, 256, 0, stream>>>(src, flow, out);
  fsplat_norm<<<nblocks, 256, 0, stream>>>(out);
}